// NormProp_46815143526639
// MI455X (gfx1250) — compile-verified
//
#include <hip/hip_runtime.h>
#include <cstddef>

typedef __attribute__((ext_vector_type(2))) float v2f;
typedef __attribute__((ext_vector_type(8))) float v8f;

#define FIN    500
#define HDIM   32
#define DDIM   64
#define LN_EPS 1e-6f

// D = A(16x4 f32) * B(4x16 f32) + C(16x16 f32), fp32 WMMA (CDNA5)
__device__ __forceinline__ v8f wmma4(v2f a, v2f b, v8f c) {
  return __builtin_amdgcn_wmma_f32_16x16x4_f32(false, a, false, b, (short)0, c,
                                               false, false);
}

__global__ __launch_bounds__(256) void zero_f32_kernel(float* __restrict__ p,
                                                       long long n) {
  long long i = (long long)blockIdx.x * blockDim.x + threadIdx.x;
  if (i < n) p[i] = 0.0f;
}

// ---------------------------------------------------------------------------
// Encoder: h0 = L2norm( relu(LN(x@W1 + b1))@W2 + b2 ), one 16-row tile / wave
// ---------------------------------------------------------------------------
__global__ __launch_bounds__(256) void encoder_kernel(
    const float* __restrict__ x, const float* __restrict__ W1,
    const float* __restrict__ b1, const float* __restrict__ gamma,
    const float* __restrict__ beta, const float* __restrict__ W2,
    const float* __restrict__ b2, float* __restrict__ h0, int N) {
  __shared__ float lds_h[8][16 * HDIM];  // per-wave 16x32 relu(LN(h)) tile

  const int wave = threadIdx.x >> 5;  // 0..7 (wave32)
  const int lane = threadIdx.x & 31;
  const int half = lane >> 4;   // 0: lanes 0-15, 1: lanes 16-31
  const int ln16 = lane & 15;   // N-index (B/C/D) or M-index (A)
  const int row0 = (blockIdx.x * 8 + wave) * 16;
  if (row0 >= N) return;  // wave-uniform: EXEC stays all-1s for WMMA below

  // ---- GEMM1: x[16, 500] @ W1[500, 32] -> two 16x16 accumulators ----------
  // fp32 A layout: lanes 0-15 hold K=k+0,k+1 ; lanes 16-31 hold K=k+2,k+3
  const int arow = min(row0 + ln16, N - 1);
  const float* __restrict__ xrow = x + (size_t)arow * FIN;
  v8f acc0 = {};
  v8f acc1 = {};
#pragma unroll 2
  for (int k = 0; k < FIN; k += 4) {
    if ((k & 31) == 0) __builtin_prefetch(xrow + k + 128, 0, 0);
    const int kk = k + 2 * half;
    v2f a;
    a.x = xrow[kk + 0];
    a.y = xrow[kk + 1];
    v2f bA, bB;  // B fragments for N-tiles [0:16) and [16:32)
    bA.x = W1[(size_t)(kk + 0) * HDIM + ln16];
    bA.y = W1[(size_t)(kk + 1) * HDIM + ln16];
    bB.x = W1[(size_t)(kk + 0) * HDIM + 16 + ln16];
    bB.y = W1[(size_t)(kk + 1) * HDIM + 16 + ln16];
    acc0 = wmma4(a, bA, acc0);
    acc1 = wmma4(a, bB, acc1);
  }

  // ---- bias + LayerNorm(H=32) + relu --------------------------------------
  // C/D layout: register j, half h  ->  row m = 8*h + j ; column n = ln16
  const float b1c0 = b1[ln16],      b1c1 = b1[16 + ln16];
  const float g0   = gamma[ln16],   g1   = gamma[16 + ln16];
  const float be0  = beta[ln16],    be1  = beta[16 + ln16];
  float* __restrict__ lh = &lds_h[wave][0];
#pragma unroll
  for (int j = 0; j < 8; ++j) {
    float v0 = acc0[j] + b1c0;
    float v1 = acc1[j] + b1c1;
    float s = v0 + v1;
    float q = v0 * v0 + v1 * v1;
#pragma unroll
    for (int m = 1; m < 16; m <<= 1) {  // stays inside the 16-lane half-group
      s += __shfl_xor(s, m, 32);
      q += __shfl_xor(q, m, 32);
    }
    const float mean = s * (1.0f / 32.0f);
    const float var  = q * (1.0f / 32.0f) - mean * mean;
    const float inv  = rsqrtf(var + LN_EPS);
    v0 = fmaxf((v0 - mean) * inv * g0 + be0, 0.0f);
    v1 = fmaxf((v1 - mean) * inv * g1 + be1, 0.0f);
    const int m = half * 8 + j;
    lh[m * HDIM + ln16]      = v0;
    lh[m * HDIM + 16 + ln16] = v1;
  }
  // Same-wave LDS RAW: DS ops are in-order per wave; compiler inserts s_wait_dscnt.

  // ---- GEMM2: relu(h)[16,32] @ W2[32,64] -> four 16x16 accumulators -------
  v8f d0 = {}, d1 = {}, d2 = {}, d3 = {};
#pragma unroll
  for (int k = 0; k < HDIM; k += 4) {
    const int kk = k + 2 * half;
    v2f a;
    a.x = lh[ln16 * HDIM + kk + 0];
    a.y = lh[ln16 * HDIM + kk + 1];
    v2f b0f, b1f, b2f, b3f;
    b0f.x = W2[(size_t)(kk + 0) * DDIM + 0 + ln16];
    b0f.y = W2[(size_t)(kk + 1) * DDIM + 0 + ln16];
    b1f.x = W2[(size_t)(kk + 0) * DDIM + 16 + ln16];
    b1f.y = W2[(size_t)(kk + 1) * DDIM + 16 + ln16];
    b2f.x = W2[(size_t)(kk + 0) * DDIM + 32 + ln16];
    b2f.y = W2[(size_t)(kk + 1) * DDIM + 32 + ln16];
    b3f.x = W2[(size_t)(kk + 0) * DDIM + 48 + ln16];
    b3f.y = W2[(size_t)(kk + 1) * DDIM + 48 + ln16];
    d0 = wmma4(a, b0f, d0);
    d1 = wmma4(a, b1f, d1);
    d2 = wmma4(a, b2f, d2);
    d3 = wmma4(a, b3f, d3);
  }

  // ---- + b2, row-wise L2 normalize, store h0 ------------------------------
  const float bb0 = b2[0 + ln16],  bb1 = b2[16 + ln16];
  const float bb2 = b2[32 + ln16], bb3 = b2[48 + ln16];
#pragma unroll
  for (int j = 0; j < 8; ++j) {
    float e0 = d0[j] + bb0;
    float e1 = d1[j] + bb1;
    float e2 = d2[j] + bb2;
    float e3 = d3[j] + bb3;
    float q = e0 * e0 + e1 * e1 + e2 * e2 + e3 * e3;
#pragma unroll
    for (int m = 1; m < 16; m <<= 1) q += __shfl_xor(q, m, 32);
    const float scale = 1.0f / fmaxf(sqrtf(q), 1e-12f);
    const int row = row0 + half * 8 + j;
    if (row < N) {
      float* __restrict__ o = h0 + (size_t)row * DDIM;
      o[0 + ln16]  = e0 * scale;
      o[16 + ln16] = e1 * scale;
      o[32 + ln16] = e2 * scale;
      o[48 + ln16] = e3 * scale;
    }
  }
}

// ---------------------------------------------------------------------------
// gcn_norm: deg[col] += 1 over all edges, then deg <- deg>0 ? rsqrt(deg) : 0
// ---------------------------------------------------------------------------
__global__ __launch_bounds__(256) void degree_kernel(const int* __restrict__ col,
                                                     float* __restrict__ deg,
                                                     int EN) {
  int e = blockIdx.x * blockDim.x + threadIdx.x;
  if (e < EN) unsafeAtomicAdd(&deg[col[e]], 1.0f);
}

__global__ __launch_bounds__(256) void dinv_kernel(float* __restrict__ deg,
                                                   int N) {
  int i = blockIdx.x * blockDim.x + threadIdx.x;
  if (i < N) {
    float d = deg[i];
    deg[i] = (d > 0.0f) ? rsqrtf(d) : 0.0f;
  }
}

// ---------------------------------------------------------------------------
// One propagation step: hdst[col] += dinv[row]*dinv[col] * hsrc[row]
// One (edge, dim) pair per lane: 64 lanes/edge -> coalesced gather + atomics.
// ---------------------------------------------------------------------------
__global__ __launch_bounds__(256) void prop_kernel(
    const float* __restrict__ hsrc, float* __restrict__ hdst,
    const int* __restrict__ row, const int* __restrict__ col,
    const float* __restrict__ dinv, long long EN) {
  long long t = (long long)blockIdx.x * 256 + threadIdx.x;
  long long e = t >> 6;
  int dd = (int)(t & 63);
  if (e < EN) {
    int r = row[e];
    int c = col[e];
    float nm = dinv[r] * dinv[c];
    float v = nm * hsrc[(size_t)r * DDIM + dd];
    unsafeAtomicAdd(&hdst[(size_t)c * DDIM + dd], v);
  }
}

extern "C" void kernel_launch(void* const* d_in, const int* in_sizes, int n_in,
                              void* d_out, int out_size, void* d_ws,
                              size_t ws_size, hipStream_t stream) {
  const float* x     = (const float*)d_in[0];
  const int*   ei    = (const int*)d_in[1];
  const float* W1    = (const float*)d_in[2];
  const float* b1    = (const float*)d_in[3];
  const float* gamma = (const float*)d_in[4];
  const float* beta  = (const float*)d_in[5];
  const float* W2    = (const float*)d_in[6];
  const float* b2    = (const float*)d_in[7];

  const int Fin = in_sizes[2] / HDIM;       // 500
  const int N   = in_sizes[0] / Fin;        // 100000
  const int EN  = in_sizes[1] / 2;          // E + N edges
  const int* erow = ei;                      // edge_index[0]
  const int* ecol = ei + EN;                 // edge_index[1]

  float* out0 = (float*)d_out;               // h0  [N, 64]
  float* out1 = out0 + (size_t)N * DDIM;     // h1
  float* out2 = out1 + (size_t)N * DDIM;     // h2
  float* deg  = (float*)d_ws;                // [N] degrees -> dinv (in place)

  // zero the accumulated outputs and the degree buffer (harness poisons them)
  {
    long long nz = 2LL * N * DDIM;
    int blocks = (int)((nz + 255) / 256);
    zero_f32_kernel<<<blocks, 256, 0, stream>>>(out1, nz);
    zero_f32_kernel<<<(N + 255) / 256, 256, 0, stream>>>(deg, (long long)N);
  }

  // encoder -> h0
  {
    int blocks = (N + 127) / 128;  // 8 waves x 16 rows per block
    encoder_kernel<<<blocks, 256, 0, stream>>>(x, W1, b1, gamma, beta, W2, b2,
                                               out0, N);
  }

  // symmetric normalization coefficients
  degree_kernel<<<(EN + 255) / 256, 256, 0, stream>>>(ecol, deg, EN);
  dinv_kernel<<<(N + 255) / 256, 256, 0, stream>>>(deg, N);

  // K = 2 propagation steps
  {
    long long work = (long long)EN * DDIM;
    int blocks = (int)((work + 255) / 256);
    prop_kernel<<<blocks, 256, 0, stream>>>(out0, out1, erow, ecol, deg,
                                            (long long)EN);
    prop_kernel<<<blocks, 256, 0, stream>>>(out1, out2, erow, ecol, deg,
                                            (long long)EN);
  }
}